// Model_19164144075140
// MI455X (gfx1250) — compile-verified
//
#include <hip/hip_runtime.h>

#define EMBD 256
#define NE   42
#define NREL 97
#define IN_D 768
#define BB   4
#define NROW (BB*NE)        // 168
#define NPIX (BB*NE*NE)     // 7056
#define KCONV (9*EMBD)      // 2304
#define MSUB 3              // m-subtiles (16 rows each) per block
#define MTILE (16*MSUB)     // 48
#define MBLK (NPIX/MTILE)   // 147

typedef __attribute__((ext_vector_type(2))) float v2f;
typedef __attribute__((ext_vector_type(8))) float v8f;

__device__ __forceinline__ v8f wmma_f32(v2f a, v2f b, v8f c) {
  // V_WMMA_F32_16X16X4_F32: D = A(16x4) * B(4x16) + C(16x16), all fp32
  return __builtin_amdgcn_wmma_f32_16x16x4_f32(false, a, false, b, (short)0, c,
                                               false, false);
}

// ---------------- Stage 1: e = entity @ W_reduce^T, plus 1/max(||e||,eps) ---
__global__ void k_reduce(const float* __restrict__ ent,
                         const float* __restrict__ Wr,
                         float* __restrict__ e, float* __restrict__ rinv) {
  __shared__ float sx[IN_D];
  __shared__ float red[256];
  const int r = blockIdx.x, t = threadIdx.x;
  for (int i = t; i < IN_D; i += 256) sx[i] = ent[r * IN_D + i];
  __syncthreads();
  float acc = 0.f;
  const float* w = Wr + t * IN_D;
  #pragma unroll 8
  for (int i = 0; i < IN_D; ++i) acc += sx[i] * w[i];
  e[r * EMBD + t] = acc;
  red[t] = acc * acc;
  __syncthreads();
  for (int s = 128; s > 0; s >>= 1) {
    if (t < s) red[t] += red[t + s];
    __syncthreads();
  }
  if (t == 0) rinv[r] = 1.f / fmaxf(sqrtf(red[0]), 1e-6f);
}

// ---------------- Stage 1b: Se = e @ W_s^T, Oe = e @ W_o^T -------------------
__global__ void k_so(const float* __restrict__ e, const float* __restrict__ Ws,
                     const float* __restrict__ Wo, float* __restrict__ Se,
                     float* __restrict__ Oe) {
  __shared__ float ev[EMBD];
  const int r = blockIdx.x, t = threadIdx.x;
  ev[t] = e[r * EMBD + t];
  __syncthreads();
  float as = 0.f, ao = 0.f;
  const float* ws = Ws + t * EMBD;
  const float* wo = Wo + t * EMBD;
  #pragma unroll 8
  for (int i = 0; i < EMBD; ++i) { as += ev[i] * ws[i]; ao += ev[i] * wo[i]; }
  Se[r * EMBD + t] = as;
  Oe[r * EMBD + t] = ao;
}

// ---------------- Stage 2a: sim1/sim2 -> fm channels 0,1 --------------------
__global__ void k_sim12(const float* __restrict__ e,
                        const float* __restrict__ rinv,
                        float* __restrict__ fm) {
  const int i = blockIdx.x, j = blockIdx.y, b = blockIdx.z, t = threadIdx.x;
  __shared__ float red[256];
  red[t] = e[(b * NE + i) * EMBD + t] * e[(b * NE + j) * EMBD + t];
  __syncthreads();
  for (int s = 128; s > 0; s >>= 1) {
    if (t < s) red[t] += red[t + s];
    __syncthreads();
  }
  if (t == 0) {
    const float s1 = red[0];
    fm[(b * 3 + 0) * NE * NE + j * NE + i] = s1;
    fm[(b * 3 + 1) * NE * NE + j * NE + i] =
        s1 * rinv[b * NE + i] * rinv[b * NE + j];
  }
}

// ---------------- Stage 2b: sim3[b,i,k] = e_bi^T W_sim3[k] e_bi -------------
// One block per (k, b): W_sim3[k] slab is streamed exactly once per batch,
// all 42 e-rows staged in LDS. Factored as sum_f e_if * (sum_e e_ie * W_ef).
__global__ void k_sim3(const float* __restrict__ e,
                       const float* __restrict__ Wsim3,
                       float* __restrict__ fm) {
  const int k = blockIdx.x, b = blockIdx.y, t = threadIdx.x;
  __shared__ float ldse[NE * EMBD];  // 42*256 floats = 43 KB
  __shared__ float red[NE];
  for (int idx = t; idx < NE * EMBD; idx += 256)
    ldse[idx] = e[(size_t)(b * NE) * EMBD + idx];
  if (t < NE) red[t] = 0.f;
  __syncthreads();

  float inner[NE];
  #pragma unroll
  for (int i = 0; i < NE; ++i) inner[i] = 0.f;
  const float* Wk = Wsim3 + (size_t)k * EMBD * EMBD;
  for (int ee = 0; ee < EMBD; ++ee) {
    const float w = Wk[(size_t)ee * EMBD + t];  // coalesced, broadcast-reused
    #pragma unroll 6
    for (int i = 0; i < NE; ++i) inner[i] += ldse[i * EMBD + ee] * w;
  }
  for (int i = 0; i < NE; ++i) {
    float contrib = inner[i] * ldse[i * EMBD + t];
    #pragma unroll
    for (int off = 16; off > 0; off >>= 1)
      contrib += __shfl_xor(contrib, off, 32);
    if ((t & 31) == 0) atomicAdd(&red[i], contrib);
  }
  __syncthreads();
  if (t < NE) fm[(b * 3 + 2) * NE * NE + k * NE + t] = red[t];
}

// ---------------- Stage 3: conv1 3->256, relu, channels-last output ---------
__global__ void k_conv1(const float* __restrict__ fm,
                        const float* __restrict__ w1,
                        const float* __restrict__ b1,
                        float* __restrict__ h1) {
  const int p = blockIdx.x, b = blockIdx.y, o = threadIdx.x;
  __shared__ float s[3][3][NE];
  for (int idx = o; idx < 3 * 3 * NE; idx += 256) {
    const int x = idx % NE, dy = (idx / NE) % 3, ci = idx / (3 * NE);
    const int row = p + dy - 1;
    s[ci][dy][x] =
        (row >= 0 && row < NE) ? fm[(b * 3 + ci) * NE * NE + row * NE + x] : 0.f;
  }
  __syncthreads();
  float w[27];
  #pragma unroll
  for (int k = 0; k < 27; ++k) w[k] = w1[o * 27 + k];
  const float bias = b1[o];
  for (int q = 0; q < NE; ++q) {
    float acc = bias;
    #pragma unroll
    for (int ci = 0; ci < 3; ++ci)
      #pragma unroll
      for (int dy = 0; dy < 3; ++dy)
        #pragma unroll
        for (int dx = 0; dx < 3; ++dx) {
          const int col = q + dx - 1;
          if (col >= 0 && col < NE)
            acc += s[ci][dy][col] * w[ci * 9 + dy * 3 + dx];
        }
    h1[((b * NE + p) * NE + q) * EMBD + o] = fmaxf(acc, 0.f);
  }
}

// ---------------- Stage 0: weight transposes for coalesced B fragments ------
__global__ void k_prep(const float* __restrict__ w2,
                       const float* __restrict__ wlin,
                       float* __restrict__ W2t, float* __restrict__ Wlt) {
  const int idx = blockIdx.x * 256 + threadIdx.x;
  if (idx < KCONV * EMBD) {  // W2t[k = dyx*256+ci][o]
    const int o = idx % EMBD, k = idx / EMBD;
    const int dyx = k >> 8, ci = k & 255;
    W2t[idx] = w2[o * KCONV + ci * 9 + dyx];
  }
  if (idx < EMBD * EMBD) {  // Wlt[c][n]
    const int n = idx % EMBD, c = idx / EMBD;
    Wlt[idx] = wlin[n * EMBD + c];
  }
}

// ---------------- Stage 4: conv2 as implicit GEMM (WMMA f32) ----------------
// M=7056 rows (b,p,q), K=2304 (k = (dy*3+dx)*256 + ci), N=256 outputs.
// 48-row M-tile: 12 WMMAs share each set of 8 B-fragment loads.
__global__ void k_conv2(const float* __restrict__ h1,
                        const float* __restrict__ W2t,
                        const float* __restrict__ c2b,
                        float* __restrict__ h2) {
  const int wave = threadIdx.x >> 5, l = threadIdx.x & 31;
  const int kh = l >> 4, ll = l & 15;
  const int rowbase = blockIdx.x * MTILE;
  int bb[MSUB], pb[MSUB], qb[MSUB];
  #pragma unroll
  for (int mt = 0; mt < MSUB; ++mt) {
    const int row = rowbase + mt * 16 + ll;
    bb[mt] = row / (NE * NE);
    const int rem = row % (NE * NE);
    pb[mt] = rem / NE;
    qb[mt] = rem % NE;
  }

  v8f acc[MSUB][4];
  #pragma unroll
  for (int mt = 0; mt < MSUB; ++mt)
    #pragma unroll
    for (int nt = 0; nt < 4; ++nt) acc[mt][nt] = (v8f)(0.f);

  const int n0 = wave * 64 + ll;
  for (int k0 = 0; k0 < KCONV; k0 += 4) {
    const int dyx = k0 >> 8;
    const int dy = dyx / 3, dx = dyx % 3;
    const int ci = (k0 & 255) + 2 * kh;
    v2f a[MSUB];
    #pragma unroll
    for (int mt = 0; mt < MSUB; ++mt) {
      const int y = pb[mt] + dy - 1, x = qb[mt] + dx - 1;
      v2f av = {0.f, 0.f};
      if (y >= 0 && y < NE && x >= 0 && x < NE)
        av = *(const v2f*)&h1[(((size_t)(bb[mt] * NE + y) * NE + x) * EMBD) + ci];
      a[mt] = av;
    }
    const int kk = k0 + 2 * kh;
    #pragma unroll
    for (int nt = 0; nt < 4; ++nt) {
      v2f bf;
      bf.x = W2t[(size_t)kk * EMBD + n0 + nt * 16];
      bf.y = W2t[(size_t)(kk + 1) * EMBD + n0 + nt * 16];
      #pragma unroll
      for (int mt = 0; mt < MSUB; ++mt)
        acc[mt][nt] = wmma_f32(a[mt], bf, acc[mt][nt]);
    }
  }
  #pragma unroll
  for (int nt = 0; nt < 4; ++nt) {
    const int col = wave * 64 + nt * 16 + ll;
    const float bias = c2b[col];
    #pragma unroll
    for (int mt = 0; mt < MSUB; ++mt)
      #pragma unroll
      for (int r = 0; r < 8; ++r) {
        const int m = mt * 16 + r + 8 * kh;
        h2[(size_t)(rowbase + m) * EMBD + col] = acc[mt][nt][r] + bias;
      }
  }
}

// ---------------- Stage 5: attn GEMM + fused tanh(zs/zo) (WMMA f32) ---------
__global__ void k_attn(const float* __restrict__ h2,
                       const float* __restrict__ Wlt,
                       const float* __restrict__ blin,
                       const float* __restrict__ Se,
                       const float* __restrict__ Oe,
                       float* __restrict__ zs, float* __restrict__ zo) {
  const int wave = threadIdx.x >> 5, l = threadIdx.x & 31;
  const int kh = l >> 4, ll = l & 15;
  const int rowbase = blockIdx.x * MTILE;

  v8f acc[MSUB][4];
  #pragma unroll
  for (int mt = 0; mt < MSUB; ++mt)
    #pragma unroll
    for (int nt = 0; nt < 4; ++nt) acc[mt][nt] = (v8f)(0.f);

  const int n0 = wave * 64 + ll;
  for (int k0 = 0; k0 < EMBD; k0 += 4) {
    const int kk = k0 + 2 * kh;
    v2f a[MSUB];
    #pragma unroll
    for (int mt = 0; mt < MSUB; ++mt)
      a[mt] = *(const v2f*)&h2[(size_t)(rowbase + mt * 16 + ll) * EMBD + kk];
    #pragma unroll
    for (int nt = 0; nt < 4; ++nt) {
      v2f bf;
      bf.x = Wlt[(size_t)kk * EMBD + n0 + nt * 16];
      bf.y = Wlt[(size_t)(kk + 1) * EMBD + n0 + nt * 16];
      #pragma unroll
      for (int mt = 0; mt < MSUB; ++mt)
        acc[mt][nt] = wmma_f32(a[mt], bf, acc[mt][nt]);
    }
  }
  #pragma unroll
  for (int nt = 0; nt < 4; ++nt) {
    const int col = wave * 64 + nt * 16 + ll;
    const float bias = blin[col];
    #pragma unroll
    for (int mt = 0; mt < MSUB; ++mt)
      #pragma unroll
      for (int r = 0; r < 8; ++r) {
        const int grow = rowbase + mt * 16 + r + 8 * kh;
        const int rs = grow / NE;                            // b*42 + I
        const int ro = (grow / (NE * NE)) * NE + grow % NE;  // b*42 + J
        const float v = acc[mt][nt][r] + bias;
        zs[(size_t)grow * EMBD + col] = tanhf(v + Se[rs * EMBD + col]);
        zo[(size_t)grow * EMBD + col] = tanhf(v + Oe[ro * EMBD + col]);
      }
  }
}

// ---------------- Stage 6: p[r,o] = zs[r] W_bil[o] zo[r]^T + b (WMMA f32) ---
__global__ void k_bil(const float* __restrict__ zs, const float* __restrict__ zo,
                      const float* __restrict__ Wbil,
                      const float* __restrict__ bbil, float* __restrict__ out) {
  const int wave = threadIdx.x >> 5, l = threadIdx.x & 31;
  const int kh = l >> 4, ll = l & 15;
  const int rowbase = blockIdx.x * MTILE;
  const int o = blockIdx.y;
  const float* W = Wbil + (size_t)o * EMBD * EMBD;

  v8f acc[MSUB][4];
  #pragma unroll
  for (int mt = 0; mt < MSUB; ++mt)
    #pragma unroll
    for (int nt = 0; nt < 4; ++nt) acc[mt][nt] = (v8f)(0.f);

  const int n0 = wave * 64 + ll;
  for (int k0 = 0; k0 < EMBD; k0 += 4) {
    const int kk = k0 + 2 * kh;
    v2f a[MSUB];
    #pragma unroll
    for (int mt = 0; mt < MSUB; ++mt)
      a[mt] = *(const v2f*)&zs[(size_t)(rowbase + mt * 16 + ll) * EMBD + kk];
    #pragma unroll
    for (int nt = 0; nt < 4; ++nt) {
      v2f bf;
      bf.x = W[(size_t)kk * EMBD + n0 + nt * 16];
      bf.y = W[(size_t)(kk + 1) * EMBD + n0 + nt * 16];
      #pragma unroll
      for (int mt = 0; mt < MSUB; ++mt)
        acc[mt][nt] = wmma_f32(a[mt], bf, acc[mt][nt]);
    }
  }
  // elementwise multiply by zo, reduce over N (columns)
  v8f c[MSUB];
  #pragma unroll
  for (int mt = 0; mt < MSUB; ++mt) c[mt] = (v8f)(0.f);
  #pragma unroll
  for (int nt = 0; nt < 4; ++nt) {
    const int col = wave * 64 + nt * 16 + ll;
    #pragma unroll
    for (int mt = 0; mt < MSUB; ++mt)
      #pragma unroll
      for (int r = 0; r < 8; ++r) {
        const int m = mt * 16 + r + 8 * kh;
        c[mt][r] += acc[mt][nt][r] * zo[(size_t)(rowbase + m) * EMBD + col];
      }
  }
  // butterfly reduce across the 16 lanes of each half (N dimension)
  #pragma unroll
  for (int off = 1; off < 16; off <<= 1)
    #pragma unroll
    for (int mt = 0; mt < MSUB; ++mt)
      #pragma unroll
      for (int r = 0; r < 8; ++r) c[mt][r] += __shfl_xor(c[mt][r], off, 16);

  __shared__ float red[MTILE];
  if (threadIdx.x < MTILE) red[threadIdx.x] = 0.f;
  __syncthreads();
  if (ll == 0) {
    #pragma unroll
    for (int mt = 0; mt < MSUB; ++mt)
      #pragma unroll
      for (int r = 0; r < 8; ++r)
        atomicAdd(&red[mt * 16 + r + 8 * kh], c[mt][r]);
  }
  __syncthreads();
  if (threadIdx.x < MTILE)
    out[(size_t)(rowbase + threadIdx.x) * NREL + o] = red[threadIdx.x] + bbil[o];
}

// ---------------- launch ----------------------------------------------------
static float* carve(float** p, size_t n) {
  float* r = *p;
  *p += ((n + 63) & ~(size_t)63);
  return r;
}

extern "C" void kernel_launch(void* const* d_in, const int* in_sizes, int n_in,
                              void* d_out, int out_size, void* d_ws,
                              size_t ws_size, hipStream_t stream) {
  (void)in_sizes; (void)n_in; (void)out_size; (void)ws_size;
  const float* ent   = (const float*)d_in[0];
  const float* Wr    = (const float*)d_in[1];
  const float* Wsim3 = (const float*)d_in[2];
  const float* c1w   = (const float*)d_in[3];
  const float* c1b   = (const float*)d_in[4];
  const float* c2w   = (const float*)d_in[5];
  const float* c2b   = (const float*)d_in[6];
  const float* Wlin  = (const float*)d_in[7];
  const float* blin  = (const float*)d_in[8];
  const float* Ws    = (const float*)d_in[9];
  const float* Wo    = (const float*)d_in[10];
  const float* Wbil  = (const float*)d_in[11];
  const float* bbil  = (const float*)d_in[12];
  float* out = (float*)d_out;

  float* w = (float*)d_ws;
  float* e    = carve(&w, (size_t)NROW * EMBD);
  float* rinv = carve(&w, NROW);
  float* Se   = carve(&w, (size_t)NROW * EMBD);
  float* Oe   = carve(&w, (size_t)NROW * EMBD);
  float* fm   = carve(&w, (size_t)BB * 3 * NE * NE);
  float* h1   = carve(&w, (size_t)NPIX * EMBD);
  float* W2t  = carve(&w, (size_t)KCONV * EMBD);
  float* Wlt  = carve(&w, (size_t)EMBD * EMBD);
  float* h2   = carve(&w, (size_t)NPIX * EMBD);
  float* zs   = carve(&w, (size_t)NPIX * EMBD);
  float* zo   = carve(&w, (size_t)NPIX * EMBD);

  k_prep<<<KCONV, 256, 0, stream>>>(c2w, Wlin, W2t, Wlt);
  k_reduce<<<NROW, 256, 0, stream>>>(ent, Wr, e, rinv);
  k_so<<<NROW, 256, 0, stream>>>(e, Ws, Wo, Se, Oe);
  k_sim12<<<dim3(NE, NE, BB), 256, 0, stream>>>(e, rinv, fm);
  k_sim3<<<dim3(NE, BB), 256, 0, stream>>>(e, Wsim3, fm);
  k_conv1<<<dim3(NE, BB), 256, 0, stream>>>(fm, c1w, c1b, h1);
  k_conv2<<<MBLK, 128, 0, stream>>>(h1, W2t, c2b, h2);
  k_attn<<<MBLK, 128, 0, stream>>>(h2, Wlt, blin, Se, Oe, zs, zo);
  k_bil<<<dim3(MBLK, NREL), 128, 0, stream>>>(zs, zo, Wbil, bbil, out);
}